// CubEclayr_69355131895902
// MI455X (gfx1250) — compile-verified
//
#include <hip/hip_runtime.h>

#define HH 224
#define WW 224
#define TR 16          // pixel rows per tile
#define ROW_TILES 14   // 14*16 = 224
#define LDSTRIDE 228   // padded row stride (floats), multiple of 4
#define NB 33          // bins 0..32 (32 = "never alive", dropped)
#define NCH 192        // 64*3 channel images

#if __has_builtin(__builtin_amdgcn_global_load_async_to_lds_b128) && __has_builtin(__builtin_amdgcn_s_wait_asynccnt)
#define USE_ASYNC 1
#else
#define USE_ASYNC 0
#endif

typedef __attribute__((ext_vector_type(2))) float v2f;
typedef __attribute__((ext_vector_type(8))) float v8f;
typedef int b128_t __attribute__((vector_size(16)));
typedef __attribute__((address_space(1))) b128_t* gptr_b128;
typedef __attribute__((address_space(3))) b128_t* lptr_b128;

__device__ __forceinline__ int binof(float f) {
    // searchsorted(linspace(0,1,32), f, left): first j with j/31 >= f
    int i = (int)ceilf(f * 31.0f);
    return i > 32 ? 32 : i;
}

__global__ void ecc_zero_kernel(int* __restrict__ gh) {
    gh[blockIdx.x * 256 + threadIdx.x] = 0;
}

__global__ __launch_bounds__(256) void ecc_hist_kernel(const float* __restrict__ x,
                                                       int* __restrict__ gh) {
    __shared__ __align__(16) float rows[(TR + 1) * LDSTRIDE];
    __shared__ int lh[NB * 32];

    const int ch = blockIdx.x;
    const int rt = blockIdx.y;
    const int r0 = rt * TR;
    const int tid = threadIdx.x;
    const int lane = tid & 31;

    for (int i = tid; i < NB * 32; i += 256) lh[i] = 0;

    const float* src = x + (size_t)ch * (HH * WW);
    const int C4 = WW / 4;  // 56 float4 chunks per row

    // Stage rows r0 .. r0+TR into LDS (sentinel 2.0 past the image bottom).
    for (int i = tid; i < (TR + 1) * C4; i += 256) {
        const int r  = i / C4;
        const int c4 = i - r * C4;
        const int gr = r0 + r;
        float* dst = &rows[r * LDSTRIDE + 4 * c4];
        if (gr < HH) {
#if USE_ASYNC
            __builtin_amdgcn_global_load_async_to_lds_b128(
                (gptr_b128)(src + (size_t)gr * WW + 4 * c4),
                (lptr_b128)dst, 0, 0);
#else
            *(float4*)dst = *(const float4*)(src + (size_t)gr * WW + 4 * c4);
#endif
        } else {
            *(float4*)dst = make_float4(2.f, 2.f, 2.f, 2.f);
        }
    }
#if USE_ASYNC
    __builtin_amdgcn_s_wait_asynccnt(0);
#endif
    __syncthreads();

    // Each lane handles a 4-pixel chunk; bins are monotone under max, so
    // edge/square bins are integer maxes of pixel bins.
    for (int i = tid; i < TR * C4; i += 256) {
        const int r  = i / C4;
        const int c4 = i - r * C4;
        const int x0 = 4 * c4;
        const float* ra = &rows[r * LDSTRIDE];
        const float* rb = &rows[(r + 1) * LDSTRIDE];
        float4 a = *(const float4*)(ra + x0);
        float4 b = *(const float4*)(rb + x0);
        const bool edge = (x0 + 4 >= WW);
        const float a4 = edge ? 2.f : ra[x0 + 4];
        const float b4 = edge ? 2.f : rb[x0 + 4];
        const int ia[5] = {binof(a.x), binof(a.y), binof(a.z), binof(a.w), binof(a4)};
        const int ib[5] = {binof(b.x), binof(b.y), binof(b.z), binof(b.w), binof(b4)};
#pragma unroll
        for (int p = 0; p < 4; ++p) {
            const int iv  = ia[p];
            const int ihe = max(ia[p], ia[p + 1]);
            const int ive = max(ia[p], ib[p]);
            const int isq = max(ihe, max(ib[p], ib[p + 1]));
            // lane-strided layout: provably zero LDS bank conflicts on wave32
            atomicAdd(&lh[iv  * 32 + lane], 1);
            atomicAdd(&lh[isq * 32 + lane], 1);
            atomicAdd(&lh[ihe * 32 + lane], -1);
            atomicAdd(&lh[ive * 32 + lane], -1);
        }
    }
    __syncthreads();

    // Fold 32 lane-private copies; bins >= 32 never become alive -> dropped.
    if (tid < 32) {
        int s = 0;
#pragma unroll
        for (int l = 0; l < 32; ++l) s += lh[tid * 32 + l];
        atomicAdd(&gh[ch * 32 + tid], s);
    }
}

// ecc[192][32] = hist[192][32] x L[32][32], L[j][t] = (j <= t), on the matrix pipe.
// V_WMMA_F32_16X16X4_F32: M=16 channels, N=16 thresholds, K=4 bins per step.
__global__ __launch_bounds__(32) void ecc_cumsum_wmma_kernel(const int* __restrict__ gh,
                                                             float* __restrict__ out) {
    const int mt  = blockIdx.x % (NCH / 16);  // channel tile (12)
    const int nt  = blockIdx.x / (NCH / 16);  // threshold tile (2)
    const int ch0 = mt * 16;
    const int t0  = nt * 16;
    const int lane  = threadIdx.x;
    const int mn    = lane & 15;          // M index for A, N index for B
    const int khalf = (lane >> 4) * 2;    // lanes 0-15: k 0,1 ; lanes 16-31: k 2,3

    v8f c = {0.f, 0.f, 0.f, 0.f, 0.f, 0.f, 0.f, 0.f};
#pragma unroll
    for (int kk = 0; kk < 32; kk += 4) {
        v2f a, b;
        a.x = (float)gh[(ch0 + mn) * 32 + kk + khalf + 0];
        a.y = (float)gh[(ch0 + mn) * 32 + kk + khalf + 1];
        b.x = (kk + khalf + 0 <= t0 + mn) ? 1.f : 0.f;
        b.y = (kk + khalf + 1 <= t0 + mn) ? 1.f : 0.f;
        c = __builtin_amdgcn_wmma_f32_16x16x4_f32(false, a, false, b,
                                                  (short)0, c, false, false);
    }
    // D layout: VGPR r holds M = r + 8*(lane>=16), N = lane&15
    const int mbase = (lane >> 4) * 8;
#pragma unroll
    for (int r = 0; r < 8; ++r) {
        out[(ch0 + mbase + r) * 32 + t0 + mn] = c[r];
    }
}

extern "C" void kernel_launch(void* const* d_in, const int* in_sizes, int n_in,
                              void* d_out, int out_size, void* d_ws, size_t ws_size,
                              hipStream_t stream) {
    (void)in_sizes; (void)n_in; (void)out_size; (void)ws_size;
    const float* x = (const float*)d_in[0];
    float* out = (float*)d_out;
    int* hist = (int*)d_ws;  // NCH*32 ints = 24 KB

    ecc_zero_kernel<<<(NCH * 32) / 256, 256, 0, stream>>>(hist);
    dim3 grid(NCH, ROW_TILES);
    ecc_hist_kernel<<<grid, 256, 0, stream>>>(x, hist);
    ecc_cumsum_wmma_kernel<<<(NCH / 16) * 2, 32, 0, stream>>>(hist, out);
}